// CurveEval_80779744903772
// MI455X (gfx1250) — compile-verified
//
#include <hip/hip_runtime.h>
#include <stdint.h>

// B-spline curve evaluation (clamped cubic), B=8192 curves, M=100 ctrl pts,
// T=512 eval points, DIM=3.  Memory-bound: ~61 MB traffic -> ~2.6 us at 23.3 TB/s.
// CDNA5 paths used: async global->LDS staging (GLOBAL_LOAD_ASYNC_TO_LDS_B128 +
// s_wait_asynccnt) of per-curve knots/ctrl points (reused 512x each), wave32
// block sizing, float4-padded LDS ctrl layout for b128 gathers, NT stores.

#define NB_M      100
#define NB_P      3
#define NB_DIM    3
#define NB_T      512
#define NB_K      (NB_M + NB_P + 1)   /* 104 */
#define THREADS   256

// Low 32 bits of a generic pointer into LDS == group-segment byte offset
// (ISA 10.2: LDS aperture maps addr[31:0] to the LDS offset).
__device__ __forceinline__ unsigned lds_off_u32(const void* p) {
  return (unsigned)(uintptr_t)p;
}

// GVS-mode async copy: each active lane copies 16 bytes from global address saddr plus voff to the LDS address in vdst.
// Tracked with ASYNCcnt (CDNA5 ISA).
__device__ __forceinline__ void async_copy_b128(unsigned lds_byte_off,
                                                unsigned gbl_byte_off,
                                                const void* sbase) {
  asm volatile("global_load_async_to_lds_b128 %0, %1, %2"
               :: "v"(lds_byte_off), "v"(gbl_byte_off), "s"(sbase)
               : "memory");
}

__device__ __forceinline__ void wait_async_zero() {
  asm volatile("s_wait_asynccnt 0" ::: "memory");
}

__global__ __launch_bounds__(THREADS) void
bspline_eval_kernel(const float* __restrict__ ctrl,   // [B,100,3]
                    const float* __restrict__ knot,   // [B,104]
                    float* __restrict__ out)          // [B,512,3]
{
  __shared__ alignas(16) float  sk [NB_K];            // 104 f32 = 26 x B128
  __shared__ alignas(16) float  scp[NB_M * NB_DIM];   // packed stage: 300 f32 = 75 x B128
  __shared__ alignas(16) float4 sc4[NB_M];            // padded ctrl, one b128 per tap

  const int b   = (int)blockIdx.x;
  const int tid = (int)threadIdx.x;

  const float* gk = knot + (size_t)b * NB_K;            // 416 B, 16B-aligned
  const float* gc = ctrl + (size_t)b * (NB_M * NB_DIM); // 1200 B, 16B-aligned

  if (tid < 26) {
    unsigned off = (unsigned)tid * 16u;
    async_copy_b128(lds_off_u32(sk) + off, off, (const void*)gk);
  } else if (tid < 26 + 75) {
    unsigned off = (unsigned)(tid - 26) * 16u;
    async_copy_b128(lds_off_u32(scp) + off, off, (const void*)gc);
  }
  wait_async_zero();   // issuing wave's deposits are in LDS
  __syncthreads();     // visible to all 8 waves

  // Repack ctrl points float3 -> float4 so each tap is one ds_load_b128.
  if (tid < NB_M) {
    float4 p;
    p.x = scp[tid * 3 + 0];
    p.y = scp[tid * 3 + 1];
    p.z = scp[tid * 3 + 2];
    p.w = 0.0f;
    sc4[tid] = p;
  }
  __syncthreads();

  const float step = (1.0f - 2e-05f) / (float)(NB_T - 1);

  #pragma unroll
  for (int rep = 0; rep < NB_T / THREADS; ++rep) {
    const int   t = tid + rep * THREADS;
    const float u = 1e-05f + (float)t * step;

    // Reference span rule: argmin over i in [0,97] of masked(u - U[3+i]),
    // masked = diff if diff > 1e-8 else 1.0, first index on ties.
    // Sorted knots => predicate (u - U[3+i] > 1e-8) is a monotone prefix;
    // span = 3 + (count-1).  count >= 1 since U[3] == 0 and u >= 1e-5.
    int lo = 0, hi = NB_M - NB_P + 1;   // 98 candidates
    while (lo < hi) {
      int mid = (lo + hi) >> 1;
      if ((u - sk[NB_P + mid]) > 1e-08f) lo = mid + 1; else hi = mid;
    }
    const int span = lo + (NB_P - 1);   // in [3, 99]

    const float km2 = sk[span - 2], km1 = sk[span - 1], k0 = sk[span];
    const float k1  = sk[span + 1], k2  = sk[span + 2], k3 = sk[span + 3];

    // Cox-de Boor, degree 3, exactly the reference recursion
    // (denom computed as (U1-u)+(u-U2); denom==0 -> temp = 1e-4).
    float N0, N1, N2, N3, saved, denom, temp;
    // k = 1
    denom = (k1 - u) + (u - k0);
    temp  = (denom == 0.0f) ? 1e-4f : (1.0f / denom);
    N0 = (k1 - u) * temp;
    N1 = (u - k0) * temp;
    // k = 2
    denom = (k1 - u) + (u - km1);
    temp  = (denom == 0.0f) ? 1e-4f : (N0 / denom);
    N0    = (k1 - u) * temp;
    saved = (u - km1) * temp;
    denom = (k2 - u) + (u - k0);
    temp  = (denom == 0.0f) ? 1e-4f : (N1 / denom);
    N1 = saved + (k2 - u) * temp;
    N2 = (u - k0) * temp;
    // k = 3
    denom = (k1 - u) + (u - km2);
    temp  = (denom == 0.0f) ? 1e-4f : (N0 / denom);
    N0    = (k1 - u) * temp;
    saved = (u - km2) * temp;
    denom = (k2 - u) + (u - km1);
    temp  = (denom == 0.0f) ? 1e-4f : (N1 / denom);
    N1    = saved + (k2 - u) * temp;
    saved = (u - km1) * temp;
    denom = (k3 - u) + (u - k0);
    temp  = (denom == 0.0f) ? 1e-4f : (N2 / denom);
    N2 = saved + (k3 - u) * temp;
    N3 = (u - k0) * temp;

    // 4-tap x 3-dim stencil: four ds_load_b128 taps
    const float4 c0 = sc4[span - 3];
    const float4 c1 = sc4[span - 2];
    const float4 c2 = sc4[span - 1];
    const float4 c3 = sc4[span - 0];
    const float x = N0 * c0.x + N1 * c1.x + N2 * c2.x + N3 * c3.x;
    const float y = N0 * c0.y + N1 * c1.y + N2 * c2.y + N3 * c3.y;
    const float z = N0 * c0.z + N1 * c1.z + N2 * c2.z + N3 * c3.z;

    float* o = out + ((size_t)b * NB_T + t) * NB_DIM;
    __builtin_nontemporal_store(x, o + 0);
    __builtin_nontemporal_store(y, o + 1);
    __builtin_nontemporal_store(z, o + 2);
  }
}

extern "C" void kernel_launch(void* const* d_in, const int* in_sizes, int n_in,
                              void* d_out, int out_size, void* d_ws, size_t ws_size,
                              hipStream_t stream) {
  (void)n_in; (void)out_size; (void)d_ws; (void)ws_size;
  const float* ctrl = (const float*)d_in[0];  // setup_inputs order: ctrl_pts, knot_u
  const float* knot = (const float*)d_in[1];
  float*       out  = (float*)d_out;

  const int B = in_sizes[0] / (NB_M * NB_DIM);   // 8192
  hipLaunchKernelGGL(bspline_eval_kernel, dim3(B), dim3(THREADS), 0, stream,
                     ctrl, knot, out);
}